// FeatureSqueezer_12137577578724
// MI455X (gfx1250) — compile-verified
//
#include <hip/hip_runtime.h>
#include <hip/hip_bf16.h>
#include <stdint.h>

// FeatureSqueezer: quantize (trunc(x*scale)/scale) + 3x3 median filter,
// reflect padding. B x 1024 x 1024 fp32.
//
// CDNA5 strategy:
//  - async global->LDS gather (GLOBAL_LOAD_ASYNC_TO_LDS_B32, ASYNCcnt) of the
//    reflect-padded 34x34 tile; 4 contiguous copies per computed address so
//    the reflect/address VALU is amortized 4x.
//  - in-LDS quantize as flat float4 with a hoisted reciprocal.
//  - exact 13-op median-of-9: per-column v_min3/v_med3/v_max3 sorts, then
//    med3(max3(lo), med3(mid), min3(hi)).

#define TILE     32
#define PT       (TILE + 2)   // 34 padded rows / cols
#define LSTRIDE  36           // floats per LDS row (keeps banks clean)
#define NTHREADS 256

#if defined(__has_builtin)
#  if __has_builtin(__builtin_amdgcn_global_load_async_to_lds_b32)
#    define USE_ASYNC_LDS 1
#  endif
#endif
#ifndef USE_ASYNC_LDS
#  define USE_ASYNC_LDS 0
#endif

// Builtin signature (from hipcc diagnostic): (global-AS int*, LDS-AS int*, Ii, Ii)
typedef __attribute__((address_space(1))) int* as1_ip;
typedef __attribute__((address_space(3))) int* as3_ip;

__device__ __forceinline__ float min3f(float a, float b, float c) {
    return fminf(fminf(a, b), c);           // -> v_min3_f32
}
__device__ __forceinline__ float max3f(float a, float b, float c) {
    return fmaxf(fmaxf(a, b), c);           // -> v_max3_f32
}
__device__ __forceinline__ float med3f(float a, float b, float c) {
#if defined(__has_builtin) && __has_builtin(__builtin_amdgcn_fmed3f)
    return __builtin_amdgcn_fmed3f(a, b, c);  // -> v_med3_f32
#else
    return fmaxf(fminf(a, b), fminf(fmaxf(a, b), c));
#endif
}

// Exact 3x3 median: sort the three columns, then combine. 13 VALU ops.
__device__ __forceinline__ float median9(float a0, float a1, float a2,
                                         float b0, float b1, float b2,
                                         float c0, float c1, float c2) {
    const float lo0 = min3f(a0, b0, c0), mi0 = med3f(a0, b0, c0), hi0 = max3f(a0, b0, c0);
    const float lo1 = min3f(a1, b1, c1), mi1 = med3f(a1, b1, c1), hi1 = max3f(a1, b1, c1);
    const float lo2 = min3f(a2, b2, c2), mi2 = med3f(a2, b2, c2), hi2 = max3f(a2, b2, c2);
    return med3f(max3f(lo0, lo1, lo2), med3f(mi0, mi1, mi2), min3f(hi0, hi1, hi2));
}

__device__ __forceinline__ int reflect_idx(int i, int n) {
    i = (i < 0) ? -i : i;                 // -1 -> 1 (reflect, edge excluded)
    i = (i >= n) ? (2 * n - 2 - i) : i;   //  n -> n-2
    return i;
}

__global__ __launch_bounds__(NTHREADS)
void feature_squeeze_median_kernel(const float* __restrict__ x,
                                   const int* __restrict__ bit_depth,
                                   float* __restrict__ out,
                                   int H, int W) {
    // Storage col 0 = left halo, 1..32 = interior, 33 = right halo.
    __shared__ __align__(16) float tile[PT][LSTRIDE];

    const int b   = blockIdx.z;
    const int y0  = blockIdx.y * TILE;
    const int x0  = blockIdx.x * TILE;
    const int tid = threadIdx.x;
    const size_t plane = (size_t)b * (size_t)H * (size_t)W;

    // gfx1250 speculative prefetch of the tile body.
    __builtin_prefetch(x + plane + (size_t)y0 * W + x0, 0, 1);

    // ---- Phase 1: reflect-indexed halo gather, global -> LDS --------------
    // 34 rows x (8 quad-tasks for the 32 interior cols + 2 edge tasks) = 340.
    for (int task = tid; task < PT * 10; task += NTHREADS) {
        const int row = task / 10;
        const int sub = task - row * 10;
        const int gy  = reflect_idx(y0 - 1 + row, H);
        const float* rowp = x + plane + (size_t)gy * W;
        if (sub < 8) {
            const float* gp = rowp + (x0 + sub * 4);     // interior: no x-reflect
            float* lp = &tile[row][1 + sub * 4];
#if USE_ASYNC_LDS
            __builtin_amdgcn_global_load_async_to_lds_b32((as1_ip)(gp + 0), (as3_ip)(lp + 0), 0, 0);
            __builtin_amdgcn_global_load_async_to_lds_b32((as1_ip)(gp + 1), (as3_ip)(lp + 1), 0, 0);
            __builtin_amdgcn_global_load_async_to_lds_b32((as1_ip)(gp + 2), (as3_ip)(lp + 2), 0, 0);
            __builtin_amdgcn_global_load_async_to_lds_b32((as1_ip)(gp + 3), (as3_ip)(lp + 3), 0, 0);
#else
            lp[0] = gp[0]; lp[1] = gp[1]; lp[2] = gp[2]; lp[3] = gp[3];
#endif
        } else {
            const int side = sub - 8;                    // 0 = left, 1 = right
            const int gx   = reflect_idx(side ? (x0 + TILE) : (x0 - 1), W);
            const int col  = side ? (TILE + 1) : 0;
#if USE_ASYNC_LDS
            __builtin_amdgcn_global_load_async_to_lds_b32(
                (as1_ip)(rowp + gx), (as3_ip)&tile[row][col], 0, 0);
#else
            tile[row][col] = rowp[gx];
#endif
        }
    }
#if USE_ASYNC_LDS
#  if defined(__has_builtin) && __has_builtin(__builtin_amdgcn_s_wait_asynccnt)
    __builtin_amdgcn_s_wait_asynccnt(0);
#  else
    asm volatile("s_wait_asynccnt 0" ::: "memory");
#  endif
#endif
    __syncthreads();

    // ---- Phase 2: quantize in LDS (flat float4 sweep) ---------------------
    // 34*36 = 1224 floats = 306 float4 (touching the 2 unused pad cols is
    // harmless: they are never read by the median phase).
    const int   bd    = *bit_depth;
    const float scale = 255.0f / (float)((1 << bd) - 1);
    const float inv   = 1.0f / scale;       // k*inv is <=1 ulp from k/scale
    float4* flat = (float4*)&tile[0][0];
    for (int idx = tid; idx < (PT * LSTRIDE) / 4; idx += NTHREADS) {
        float4 v = flat[idx];
        v.x = truncf(v.x * scale) * inv;
        v.y = truncf(v.y * scale) * inv;
        v.z = truncf(v.z * scale) * inv;
        v.w = truncf(v.w * scale) * inv;
        flat[idx] = v;
    }
    __syncthreads();

    // ---- Phase 3: median with rolling 3-row window ------------------------
    const int tx  = tid & (TILE - 1);   // 0..31: one wave32 per row-group
    const int ty0 = (tid >> 5) * 4;     // 4 consecutive output rows per thread

    float a0 = tile[ty0 + 0][tx], a1 = tile[ty0 + 0][tx + 1], a2 = tile[ty0 + 0][tx + 2];
    float b0 = tile[ty0 + 1][tx], b1 = tile[ty0 + 1][tx + 1], b2 = tile[ty0 + 1][tx + 2];
    float c0 = tile[ty0 + 2][tx], c1 = tile[ty0 + 2][tx + 1], c2 = tile[ty0 + 2][tx + 2];

#pragma unroll
    for (int i = 0; i < 4; ++i) {
        const float m = median9(a0, a1, a2, b0, b1, b2, c0, c1, c2);
        out[plane + (size_t)(y0 + ty0 + i) * W + (x0 + tx)] = m;
        if (i < 3) {
            a0 = b0; a1 = b1; a2 = b2;
            b0 = c0; b1 = c1; b2 = c2;
            c0 = tile[ty0 + 3 + i][tx];
            c1 = tile[ty0 + 3 + i][tx + 1];
            c2 = tile[ty0 + 3 + i][tx + 2];
        }
    }
}

extern "C" void kernel_launch(void* const* d_in, const int* in_sizes, int n_in,
                              void* d_out, int out_size, void* d_ws, size_t ws_size,
                              hipStream_t stream) {
    (void)n_in; (void)d_ws; (void)ws_size; (void)out_size;
    const float* x  = (const float*)d_in[0];
    const int*   bd = (const int*)d_in[1];
    float*       o  = (float*)d_out;

    const int H = 1024, W = 1024;
    const int B = in_sizes[0] / (H * W);

    dim3 grid(W / TILE, H / TILE, B);
    feature_squeeze_median_kernel<<<grid, NTHREADS, 0, stream>>>(x, bd, o, H, W);
}